// GraphEncoder_25305947308265
// MI455X (gfx1250) — compile-verified
//
#include <hip/hip_runtime.h>
#include <hip/hip_bf16.h>
#include <math.h>

typedef __attribute__((ext_vector_type(2))) float v2f;
typedef __attribute__((ext_vector_type(8))) float v8f;

#define H 128
#define FIN 64

__device__ __forceinline__ v8f wmma_f32(v2f a, v2f b, v8f c) {
    // D(16x16,f32) += A(16x4,f32) * B(4x16,f32)
    return __builtin_amdgcn_wmma_f32_16x16x4_f32(
        false, a, false, b, (short)0, c, false, false);
}

// ---------------------------------------------------------------------------
// Repack a K x 128 (row-major, k-major) weight matrix into the per-lane WMMA
// B-fragment layout: Wp[k0/4][nt][lane] = float2{ W[k0+koff][col], W[k0+koff+1][col] }
// with col = nt*16 + (lane&15), koff = (lane>>4)*2.  Makes the GEMM inner loop
// B loads contiguous float2 -> coalesced global_load_b64.
// ---------------------------------------------------------------------------
__global__ void k_repack(const float* __restrict__ W, float2* __restrict__ Wp, int K) {
    int i = blockIdx.x * blockDim.x + threadIdx.x;
    int total = (K / 4) * 8 * 32;
    if (i >= total) return;
    int lane = i & 31;
    int nt   = (i >> 5) & 7;
    int ks   = i >> 8;
    int c    = lane & 15;
    int koff = (lane >> 4) * 2;
    int k    = ks * 4 + koff;
    int col  = nt * 16 + c;
    Wp[i] = make_float2(W[k * H + col], W[(k + 1) * H + col]);
}

// ---------------------------------------------------------------------------
// Input projection: Y[BN,128] = relu(X[BN,64] @ W + bias), packed weights.
// One wave per 16-row tile; 8 n-tiles of 16 cols each.
// ---------------------------------------------------------------------------
__global__ __launch_bounds__(32) void k_in_gemm(
    const float* __restrict__ X, const float2* __restrict__ Wp,
    const float* __restrict__ bias, float* __restrict__ Y) {
    __shared__ float smem[16 * H];
    int tile = blockIdx.x;
    int lane = threadIdx.x;
    int row0 = tile * 16;
    int c    = lane & 15;          // column within n-tile / A row (M)
    int koff = (lane >> 4) * 2;    // lane-half selects K pair

    v8f acc[8];
    for (int nt = 0; nt < 8; ++nt) {
        float b = bias[nt * 16 + c];
        v8f z;
        for (int r = 0; r < 8; ++r) z[r] = b;
        acc[nt] = z;
    }
    for (int k0 = 0; k0 < FIN; k0 += 4) {
        v2f a;
        a.x = X[(row0 + c) * FIN + k0 + koff];
        a.y = X[(row0 + c) * FIN + k0 + koff + 1];
        const float2* wb = Wp + (size_t)(k0 >> 2) * 256 + lane;
        for (int nt = 0; nt < 8; ++nt) {
            float2 w = wb[nt * 32];
            v2f bb;
            bb.x = w.x;
            bb.y = w.y;
            acc[nt] = wmma_f32(a, bb, acc[nt]);
        }
    }
    int rbase = (lane >> 4) * 8;   // C layout: lane-half selects rows 0-7 / 8-15
    for (int nt = 0; nt < 8; ++nt)
        for (int r = 0; r < 8; ++r)
            smem[(rbase + r) * H + nt * 16 + c] = fmaxf(acc[nt][r], 0.f);
    __syncthreads();
    float4* yo = (float4*)(Y + (size_t)row0 * H);
    const float4* sm4 = (const float4*)smem;
    for (int i = 0; i < 16; ++i) {
        int f4 = i * 32 + lane;
        yo[f4] = sm4[f4];
    }
}

// ---------------------------------------------------------------------------
// Zero scratch (sums + counts), grid-stride.
// ---------------------------------------------------------------------------
__global__ void k_zero(float* __restrict__ p, int n) {
    int i = blockIdx.x * blockDim.x + threadIdx.x;
    int stride = gridDim.x * blockDim.x;
    for (; i < n; i += stride) p[i] = 0.f;
}

// ---------------------------------------------------------------------------
// Edge scatter: one wave per edge. float4 gather, 4 float atomics per lane.
// ---------------------------------------------------------------------------
__global__ __launch_bounds__(256) void k_scatter(
    const float* __restrict__ x, const int* __restrict__ ei,
    const float* __restrict__ emask, float* __restrict__ sums,
    float* __restrict__ cnt, int B_, int N_, int E_) {
    int gid  = blockIdx.x * blockDim.x + threadIdx.x;
    int wave = gid >> 5;
    int lane = gid & 31;
    if (wave >= B_ * E_) return;
    int b = wave / E_;
    int e = wave - b * E_;
    float m = emask[b * E_ + e];
    if (m <= 0.f) return;
    int src = ei[(b * 2 + 0) * E_ + e];
    int tgt = ei[(b * 2 + 1) * E_ + e];
    const float* xs = x + (size_t)(b * N_ + src) * H;
    float* sd       = sums + (size_t)(b * N_ + tgt) * H;
    float4 xv = *(const float4*)(xs + lane * 4);
    atomicAdd(sd + lane * 4 + 0, xv.x * m);
    atomicAdd(sd + lane * 4 + 1, xv.y * m);
    atomicAdd(sd + lane * 4 + 2, xv.z * m);
    atomicAdd(sd + lane * 4 + 3, xv.w * m);
    if (lane == 0) atomicAdd(cnt + b * N_ + tgt, m);
}

// ---------------------------------------------------------------------------
// One SAGE layer: y = mask * LN(relu(x@Ws + bs + (sums/cnt)@Wn + bn))
// One wave per 16-node tile; both GEMMs accumulate into the same C tile.
// Weights come pre-packed (k_repack layout).
// ---------------------------------------------------------------------------
__global__ __launch_bounds__(32) void k_layer(
    const float* __restrict__ x, const float* __restrict__ sums,
    const float* __restrict__ cnt,
    const float2* __restrict__ Wsp, const float* __restrict__ bs,
    const float2* __restrict__ Wnp, const float* __restrict__ bn,
    const float* __restrict__ lg, const float* __restrict__ lb,
    const float* __restrict__ nmask, float* __restrict__ y) {
    __shared__ float smem[16 * H];
    __shared__ float s_rcp[16];
    __shared__ float s_mu[16];
    __shared__ float s_rs[16];

    int tile = blockIdx.x;
    int lane = threadIdx.x;
    int row0 = tile * 16;
    int c    = lane & 15;
    int koff = (lane >> 4) * 2;

    if (lane < 16) s_rcp[lane] = 1.0f / fmaxf(cnt[row0 + lane], 1.0f);
    __syncthreads();
    float rcp_c = s_rcp[c];

    v8f acc[8];
    for (int nt = 0; nt < 8; ++nt) {
        float b = bs[nt * 16 + c] + bn[nt * 16 + c];
        v8f z;
        for (int r = 0; r < 8; ++r) z[r] = b;
        acc[nt] = z;
    }
    // self GEMM
    for (int k0 = 0; k0 < H; k0 += 4) {
        v2f a;
        a.x = x[(row0 + c) * H + k0 + koff];
        a.y = x[(row0 + c) * H + k0 + koff + 1];
        const float2* wb = Wsp + (size_t)(k0 >> 2) * 256 + lane;
        for (int nt = 0; nt < 8; ++nt) {
            float2 w = wb[nt * 32];
            v2f bb;
            bb.x = w.x;
            bb.y = w.y;
            acc[nt] = wmma_f32(a, bb, acc[nt]);
        }
    }
    // neighbor GEMM (mean folded into A-operand load)
    for (int k0 = 0; k0 < H; k0 += 4) {
        v2f a;
        a.x = sums[(row0 + c) * H + k0 + koff] * rcp_c;
        a.y = sums[(row0 + c) * H + k0 + koff + 1] * rcp_c;
        const float2* wb = Wnp + (size_t)(k0 >> 2) * 256 + lane;
        for (int nt = 0; nt < 8; ++nt) {
            float2 w = wb[nt * 32];
            v2f bb;
            bb.x = w.x;
            bb.y = w.y;
            acc[nt] = wmma_f32(a, bb, acc[nt]);
        }
    }
    // relu -> LDS
    int rbase = (lane >> 4) * 8;
    for (int nt = 0; nt < 8; ++nt)
        for (int r = 0; r < 8; ++r)
            smem[(rbase + r) * H + nt * 16 + c] = fmaxf(acc[nt][r], 0.f);
    __syncthreads();
    // layernorm stats: 2 lanes per row, float4 LDS reads, combine via shuffle
    {
        int rrow = lane & 15;
        int half = lane >> 4;
        const float4* sm4 = (const float4*)(smem + rrow * H + half * 64);
        float s = 0.f;
        for (int i = 0; i < 16; ++i) {
            float4 v = sm4[i];
            s += v.x + v.y + v.z + v.w;
        }
        s += __shfl_xor(s, 16);
        float mu = s * (1.0f / H);
        float var = 0.f;
        for (int i = 0; i < 16; ++i) {
            float4 v = sm4[i];
            float a0 = v.x - mu, a1 = v.y - mu, a2 = v.z - mu, a3 = v.w - mu;
            var += a0 * a0 + a1 * a1 + a2 * a2 + a3 * a3;
        }
        var += __shfl_xor(var, 16);
        if (lane < 16) {
            s_mu[lane] = mu;
            s_rs[lane] = rsqrtf(var * (1.0f / H) + 1e-5f);
        }
    }
    __syncthreads();
    // normalize, affine, node-mask; coalesced float4 stores
    float4* yo = (float4*)(y + (size_t)row0 * H);
    const float4* sm4 = (const float4*)smem;
    for (int i = 0; i < 16; ++i) {
        int f4   = i * 32 + lane;
        int flat = f4 * 4;
        int row  = flat >> 7;
        int col  = flat & (H - 1);
        float4 v = sm4[f4];
        float mu = s_mu[row], rs = s_rs[row], nm = nmask[row0 + row];
        float4 o;
        o.x = ((v.x - mu) * rs * lg[col + 0] + lb[col + 0]) * nm;
        o.y = ((v.y - mu) * rs * lg[col + 1] + lb[col + 1]) * nm;
        o.z = ((v.z - mu) * rs * lg[col + 2] + lb[col + 2]) * nm;
        o.w = ((v.w - mu) * rs * lg[col + 3] + lb[col + 3]) * nm;
        yo[f4] = o;
    }
}

// ---------------------------------------------------------------------------
// Two-stage mean/max pool over nodes.
// ---------------------------------------------------------------------------
__global__ __launch_bounds__(128) void k_pool_part(
    const float* __restrict__ x, const float* __restrict__ nmask,
    float* __restrict__ psum, float* __restrict__ pmax, float* __restrict__ pnv,
    int N_, int chunks) {
    int b = blockIdx.x, ch = blockIdx.y, d = threadIdx.x;
    int per = N_ / chunks;
    int n0  = ch * per;
    float s = 0.f, mx = -INFINITY, nv = 0.f;
    for (int i = 0; i < per; ++i) {
        int n   = n0 + i;
        float m = nmask[b * N_ + n];
        float v = x[(size_t)(b * N_ + n) * H + d] * m;
        s += v;
        nv += m;
        mx = (m > 0.f) ? fmaxf(mx, v) : mx;
    }
    int pi = b * chunks + ch;
    psum[pi * H + d] = s;
    pmax[pi * H + d] = mx;
    if (d == 0) pnv[pi] = nv;
}

__global__ __launch_bounds__(128) void k_pool_final(
    const float* __restrict__ psum, const float* __restrict__ pmax,
    const float* __restrict__ pnv, float* __restrict__ pooled, int chunks) {
    int b = blockIdx.x, d = threadIdx.x;
    float s = 0.f, mx = -INFINITY, nv = 0.f;
    for (int ch = 0; ch < chunks; ++ch) {
        int pi = b * chunks + ch;
        s += psum[pi * H + d];
        mx = fmaxf(mx, pmax[pi * H + d]);
        nv += pnv[pi];
    }
    pooled[b * 2 * H + d]     = s / fmaxf(nv, 1.f);
    pooled[b * 2 * H + H + d] = mx;
}

// ---------------------------------------------------------------------------
// MLP head (tiny, B=8 rows).
// ---------------------------------------------------------------------------
__global__ __launch_bounds__(128) void k_mlp1(
    const float* __restrict__ g, const float* __restrict__ W,
    const float* __restrict__ bias, float* __restrict__ hid) {
    int b = blockIdx.x, j = threadIdx.x;
    float acc = bias[j];
    for (int k = 0; k < 2 * H; ++k) acc += g[b * 2 * H + k] * W[k * H + j];
    hid[b * H + j] = fmaxf(acc, 0.f);
}

__global__ __launch_bounds__(256) void k_mlp2(
    const float* __restrict__ hid, const float* __restrict__ W,
    const float* __restrict__ bias, float* __restrict__ out) {
    int b = blockIdx.x, j = threadIdx.x;
    float acc = bias[j];
    for (int k = 0; k < H; ++k) acc += hid[b * H + k] * W[k * 256 + j];
    out[b * 256 + j] = acc;
}

// ---------------------------------------------------------------------------
extern "C" void kernel_launch(void* const* d_in, const int* in_sizes, int n_in,
                              void* d_out, int out_size, void* d_ws, size_t ws_size,
                              hipStream_t stream) {
    (void)in_sizes; (void)n_in; (void)out_size; (void)ws_size;
    const float* node_feats = (const float*)d_in[0];
    const int*   ei         = (const int*)d_in[1];
    const float* nmask      = (const float*)d_in[2];
    const float* emask      = (const float*)d_in[3];
    const float* in_W       = (const float*)d_in[4];
    const float* in_b       = (const float*)d_in[5];
    const float* self_W     = (const float*)d_in[6];
    const float* self_b     = (const float*)d_in[7];
    const float* neigh_W    = (const float*)d_in[8];
    const float* neigh_b    = (const float*)d_in[9];
    const float* ln_g       = (const float*)d_in[10];
    const float* ln_b       = (const float*)d_in[11];
    const float* out1_W     = (const float*)d_in[12];
    const float* out1_b     = (const float*)d_in[13];
    const float* out2_W     = (const float*)d_in[14];
    const float* out2_b     = (const float*)d_in[15];

    const int B = 8, N = 8192, E = 65536;
    const int BN = B * N;
    const int CH = 64;

    float* ws     = (float*)d_ws;
    float* xA     = ws;                           // BN*H
    float* xB     = xA + (size_t)BN * H;          // BN*H
    float* sums   = xB + (size_t)BN * H;          // BN*H
    float* cnt    = sums + (size_t)BN * H;        // BN   (contiguous with sums)
    float* pooled = cnt + BN;                     // B*2H
    float* hid    = pooled + B * 2 * H;           // B*H
    float* psum   = hid + B * H;                  // B*CH*H
    float* pmax   = psum + (size_t)B * CH * H;    // B*CH*H
    float* pnv    = pmax + (size_t)B * CH * H;    // B*CH
    float* wpack  = pnv + B * CH;                 // packed weights
    float2* in_Wp = (float2*)wpack;               // FIN*H floats = FIN*64 float2
    float2* sWp   = in_Wp + FIN * 64;             // 3 * H*64 float2
    float2* nWp   = sWp + 3 * H * 64;             // 3 * H*64 float2

    // Pack weights into per-lane WMMA B-fragment order (coalesced b64 loads).
    k_repack<<<16, 256, 0, stream>>>(in_W, in_Wp, FIN);
    for (int l = 0; l < 3; ++l) {
        k_repack<<<32, 256, 0, stream>>>(self_W + (size_t)l * H * H, sWp + (size_t)l * H * 64, H);
        k_repack<<<32, 256, 0, stream>>>(neigh_W + (size_t)l * H * H, nWp + (size_t)l * H * 64, H);
    }

    k_in_gemm<<<BN / 16, 32, 0, stream>>>(node_feats, in_Wp, in_b, xA);

    float* xin = xA;
    float* xout = xB;
    for (int l = 0; l < 3; ++l) {
        k_zero<<<4096, 256, 0, stream>>>(sums, BN * H + BN);
        k_scatter<<<(B * E * 32) / 256, 256, 0, stream>>>(xin, ei, emask, sums, cnt, B, N, E);
        k_layer<<<BN / 16, 32, 0, stream>>>(
            xin, sums, cnt,
            sWp + (size_t)l * H * 64, self_b + l * H,
            nWp + (size_t)l * H * 64, neigh_b + l * H,
            ln_g + l * H, ln_b + l * H, nmask, xout);
        float* t = xin; xin = xout; xout = t;
    }

    k_pool_part<<<dim3(B, CH), 128, 0, stream>>>(xin, nmask, psum, pmax, pnv, N, CH);
    k_pool_final<<<B, 128, 0, stream>>>(psum, pmax, pnv, pooled, CH);
    k_mlp1<<<B, 128, 0, stream>>>(pooled, out1_W, out1_b, hid);
    k_mlp2<<<B, 256, 0, stream>>>(hid, out2_W, out2_b, (float*)d_out);
}